// VectorizedMACE_49383533969494
// MI455X (gfx1250) — compile-verified
//
#include <hip/hip_runtime.h>
#include <math.h>

// ---------------------------------------------------------------------------
// Problem constants (match reference)
// ---------------------------------------------------------------------------
#define NNODES  10000
#define NEDGES  128000
#define NGRAPHS 50
#define NELEM   10
#define HID     64
#define NB      8
#define NSH     16
#define RCUT    5.0f

// d_out layout (floats): te[50] | node_e[10000] | contrib[50*4] | forces[10000*3] | nf[10000*128]
#define OUT_TE      0
#define OUT_NE      50
#define OUT_CONTRIB 10050
#define OUT_F       10250
#define OUT_NF      40250
#define OUT_TOTAL   1320250

typedef float v2f __attribute__((ext_vector_type(2)));
typedef float v8f __attribute__((ext_vector_type(8)));

// fast sigmoid: v_exp_f32 + v_rcp_f32 (avoids IEEE div_scale/div_fixup sequence)
__device__ __forceinline__ float sigmoidf_(float x)
{
    return __builtin_amdgcn_rcpf(1.f + __expf(-x));
}
__device__ __forceinline__ float siluf_(float x)    { return x * sigmoidf_(x); }
__device__ __forceinline__ float silugradf_(float x){ float s = sigmoidf_(x); return s * (1.f + x * (1.f - s)); }

// ---------------------------------------------------------------------------
// fp32 GEMM on the WMMA pipe: C[M,N] (+)= act( A[M,K] * B[K,N] )
//   - one wave32 computes a 16 x (16*NT) output strip; V_WMMA_F32_16X16X4_F32
//   - fully specialized at compile time: straight-line loads + WMMA, no branches
//   - M must be a multiple of 16 (true for all call sites: 10000 / 128000)
//   - A element (row,k) at A[row*lda + k*INCA]  (INCA=16 reads m[:,:,0] slices)
//   - TRANSB reads B^T (B stored [N,K] row-major) for all backward GEMMs
//   - GUARDN: clamp B column reads and guard stores (only the N=8 g_ef GEMM)
// ---------------------------------------------------------------------------
template<int TRANSB, int ACT, int BETA, int NT, int INCA, int GUARDN, int KC>
__global__ void k_gemm_t(const float* __restrict__ A, int lda,
                         const float* __restrict__ B, int ldb,
                         float* __restrict__ C, int ldc, int M, int N)
{
    int tile = blockIdx.x * (blockDim.x >> 5) + (threadIdx.x >> 5);
    int tilesN = (N + 16 * NT - 1) / (16 * NT);
    int tilesM = M >> 4;
    if (tile >= tilesM * tilesN) return;           // wave-uniform exit (EXEC stays full)
    int tm = tile / tilesN;
    int tn = tile - tm * tilesN;

    int lane = threadIdx.x & 31;
    int half = lane >> 4;                          // 0: K pair {0,1}, 1: K pair {2,3}
    int l15  = lane & 15;

    const float* Ap = A + (size_t)(tm * 16 + l15) * lda;
    int bc[NT];
#pragma unroll
    for (int t = 0; t < NT; ++t) {
        int c = tn * 16 * NT + t * 16 + l15;
        bc[t] = GUARDN ? (c < N ? c : N - 1) : c;  // clamp (no EXEC games)
    }

    v8f acc[NT];
    v8f zero = {};
#pragma unroll
    for (int t = 0; t < NT; ++t) acc[t] = zero;

#pragma unroll
    for (int k0 = 0; k0 < KC; k0 += 4) {
        int ka = k0 + half * 2;
        v2f a;
        if (INCA == 1) {
            a = *(const v2f*)(Ap + ka);            // b64 load
        } else {
            a.x = Ap[(size_t)ka * INCA];
            a.y = Ap[(size_t)(ka + 1) * INCA];
        }
#pragma unroll
        for (int t = 0; t < NT; ++t) {
            v2f b;
            if (TRANSB) {
                b = *(const v2f*)(B + (size_t)bc[t] * ldb + ka);   // b64 load
            } else {
                b.x = B[(size_t)(ka    ) * ldb + bc[t]];
                b.y = B[(size_t)(ka + 1) * ldb + bc[t]];
            }
            acc[t] = __builtin_amdgcn_wmma_f32_16x16x4_f32(false, a, false, b,
                                                           (short)0, acc[t], false, false);
        }
    }

#pragma unroll
    for (int t = 0; t < NT; ++t) {
#pragma unroll
        for (int v = 0; v < 8; ++v) {
            int row = tm * 16 + v + half * 8;
            int col = tn * 16 * NT + t * 16 + l15;
            if (!GUARDN || col < N) {
                size_t off = (size_t)row * ldc + col;
                float val = acc[t][v];
                if (BETA) val += C[off];
                if (ACT == 1) val = siluf_(val);
                C[off] = val;
            }
        }
    }
}

// ---------------------------------------------------------------------------
// Elementwise / graph kernels
// ---------------------------------------------------------------------------
__global__ void k_zero(float* p, int n)
{
    int i = blockIdx.x * blockDim.x + threadIdx.x;
    if (i < n) p[i] = 0.f;
}

__global__ void k_node_init(const float* __restrict__ attrs, const float* __restrict__ ae,
                            const float* __restrict__ Wemb, const int* __restrict__ batch,
                            float* __restrict__ nf0, float* __restrict__ elem,
                            float* __restrict__ node_e, float* __restrict__ contrib)
{
    int idx = blockIdx.x * blockDim.x + threadIdx.x;
    if (idx >= NNODES * HID) return;
    int n = idx >> 6, h = idx & 63;
    float acc = 0.f;
#pragma unroll
    for (int z = 0; z < NELEM; ++z) acc += attrs[n * NELEM + z] * Wemb[z * HID + h];
    nf0[idx] = acc;
    if (h == 0) {
        int zi = 0; float e0 = 0.f;
#pragma unroll
        for (int z = 0; z < NELEM; ++z) {
            float a = attrs[n * NELEM + z];
            e0 += a * ae[z];
            if (a > 0.5f) zi = z;
        }
        elem[n] = (float)zi;
        node_e[n] = e0;
        atomicAdd(&contrib[batch[n] * 4 + 0], e0);
    }
}

__global__ void k_edge_geom(const float* __restrict__ pos, const float* __restrict__ shifts,
                            const int* __restrict__ eidx,
                            float* __restrict__ ef, float* __restrict__ Ybuf)
{
    int e = blockIdx.x * blockDim.x + threadIdx.x;
    if (e >= NEDGES) return;
    int s = eidx[e], r = eidx[NEDGES + e];
    float vx = pos[r * 3 + 0] - pos[s * 3 + 0] + shifts[e * 3 + 0];
    float vy = pos[r * 3 + 1] - pos[s * 3 + 1] + shifts[e * 3 + 1];
    float vz = pos[r * 3 + 2] - pos[s * 3 + 2] + shifts[e * 3 + 2];
    float rl = sqrtf(vx * vx + vy * vy + vz * vz);
    float inv = 1.f / (rl + 1e-9f);
    float x = vx * inv, y = vy * inv, z = vz * inv;
    float x2 = x * x, y2 = y * y, z2 = z * z;
    const float s3 = 1.7320508f;
    float* Ye = Ybuf + (size_t)e * NSH;
    Ye[0] = 1.f; Ye[1] = x; Ye[2] = y; Ye[3] = z;
    Ye[4] = s3 * x * y; Ye[5] = s3 * y * z; Ye[6] = 0.5f * (3.f * z2 - 1.f);
    Ye[7] = s3 * x * z; Ye[8] = 0.5f * s3 * (x2 - y2);
    Ye[9]  = 0.7906f * y * (3.f * x2 - y2);
    Ye[10] = 3.873f * x * y * z;
    Ye[11] = 0.6124f * y * (4.f * z2 - x2 - y2);
    Ye[12] = 0.5f * z * (2.f * z2 - 3.f * x2 - 3.f * y2);
    Ye[13] = 0.6124f * x * (4.f * z2 - x2 - y2);
    Ye[14] = 1.9365f * z * (x2 - y2);
    Ye[15] = 0.7906f * x * (x2 - 3.f * y2);
    // Bessel * polynomial cutoff (p=6)
    float uc = fminf(rl * (1.f / RCUT), 1.f);
    float u2 = uc * uc, u4 = u2 * u2, u6 = u4 * u2, u7 = u6 * uc, u8 = u7 * uc;
    float fc = 1.f - 28.f * u6 + 48.f * u7 - 21.f * u8;
    if (rl >= RCUT) fc = 0.f;
    const float Cb = 0.63245553f;                 // sqrt(2/RCUT)
    const float PI_ = 3.14159265358979f;
    float* efe = ef + (size_t)e * NB;
#pragma unroll
    for (int nB = 1; nB <= NB; ++nB)
        efe[nB - 1] = Cb * __sinf((float)nB * PI_ * rl * (1.f / RCUT)) * inv * fc;
}

// m[recv,h,k] += h[send,h]*R[e,h]*Y[e,k]  (one thread per (e,h))
__global__ void k_edge_msg(const int* __restrict__ eidx, const float* __restrict__ hbuf,
                           const float* __restrict__ Rbuf, const float* __restrict__ Ybuf,
                           float* __restrict__ m)
{
    int idx = blockIdx.x * blockDim.x + threadIdx.x;
    if (idx >= NEDGES * HID) return;
    int e = idx >> 6, h = idx & 63;
    int s = eidx[e], r = eidx[NEDGES + e];
    float w = hbuf[s * HID + h] * Rbuf[(size_t)e * HID + h];
    const float* Ye = Ybuf + (size_t)e * NSH;
    float* md = m + (size_t)r * HID * NSH + h * NSH;
#pragma unroll
    for (int k = 0; k < NSH; ++k) atomicAdd(&md[k], w * Ye[k]);
}

__global__ void k_node_q(const float* __restrict__ m, float* __restrict__ q)
{
    int idx = blockIdx.x * blockDim.x + threadIdx.x;
    if (idx >= NNODES * HID) return;
    const float* mr = m + (size_t)idx * NSH;
    float acc = 0.f;
#pragma unroll
    for (int k = 0; k < NSH; ++k) acc += mr[k] * mr[k];
    q[idx] = acc;
}

// sc[n,k] = sum_h Wsc[z(n),h,k]*nf_in[n,h]  (written as base into nf_out)
__global__ void k_sc_fwd(const float* __restrict__ elem, const float* __restrict__ Wsc,
                         const float* __restrict__ nfin, float* __restrict__ nfout)
{
    int idx = blockIdx.x * blockDim.x + threadIdx.x;
    if (idx >= NNODES * HID) return;
    int n = idx >> 6, k = idx & 63;
    int z = (int)elem[n];
    const float* W = Wsc + (size_t)z * HID * HID;
    const float* nf = nfin + (size_t)n * HID;
    float acc = 0.f;
    for (int h = 0; h < HID; ++h) acc += W[h * HID + k] * nf[h];
    nfout[idx] = acc;
}

__global__ void k_readout0(const float* __restrict__ nf, const float* __restrict__ ro0,
                           const int* __restrict__ batch,
                           float* __restrict__ node_e, float* __restrict__ contrib)
{
    int n = blockIdx.x * blockDim.x + threadIdx.x;
    if (n >= NNODES) return;
    float acc = 0.f;
    for (int h = 0; h < HID; ++h) acc += nf[(size_t)n * HID + h] * ro0[h];
    node_e[n] += acc;
    atomicAdd(&contrib[batch[n] * 4 + 2], acc);
}

__global__ void k_readout1(const float* __restrict__ p, const float* __restrict__ ro1b,
                           const int* __restrict__ batch,
                           float* __restrict__ node_e, float* __restrict__ contrib)
{
    int n = blockIdx.x * blockDim.x + threadIdx.x;
    if (n >= NNODES) return;
    float acc = 0.f;
#pragma unroll
    for (int k = 0; k < 16; ++k) acc += siluf_(p[(size_t)n * 16 + k]) * ro1b[k];
    node_e[n] += acc;
    atomicAdd(&contrib[batch[n] * 4 + 3], acc);
}

__global__ void k_total(const float* __restrict__ contrib, float* __restrict__ te)
{
    int g = blockIdx.x * blockDim.x + threadIdx.x;
    if (g >= NGRAPHS) return;
    te[g] = contrib[g * 4 + 0] + contrib[g * 4 + 1] + contrib[g * 4 + 2] + contrib[g * 4 + 3];
}

__global__ void k_pack_nf(const float* __restrict__ nf1, const float* __restrict__ nf2,
                          float* __restrict__ out)
{
    int idx = blockIdx.x * blockDim.x + threadIdx.x;
    if (idx >= NNODES * 128) return;
    int n = idx >> 7, j = idx & 127;
    out[idx] = (j < HID) ? nf1[(size_t)n * HID + j] : nf2[(size_t)n * HID + (j - HID)];
}

// ------------------------------- backward ---------------------------------
__global__ void k_gp(const float* __restrict__ p, const float* __restrict__ ro1b,
                     float* __restrict__ gp)
{
    int idx = blockIdx.x * blockDim.x + threadIdx.x;
    if (idx >= NNODES * 16) return;
    gp[idx] = silugradf_(p[idx]) * ro1b[idx & 15];
}

__global__ void k_gm(const float* __restrict__ A1, const float* __restrict__ A2,
                     const float* __restrict__ m, float* __restrict__ gm)
{
    int idx = blockIdx.x * blockDim.x + threadIdx.x;
    if (idx >= NNODES * HID) return;
    float a1 = A1[idx], a2 = A2[idx];
    const float* mr = m + (size_t)idx * NSH;
    float* gr = gm + (size_t)idx * NSH;
    gr[0] = a1 + 2.f * mr[0] * a2;
#pragma unroll
    for (int k = 1; k < NSH; ++k) gr[k] = 2.f * mr[k] * a2;
}

// one wave32 per edge: gR[e,h], gY[e,k], (opt) gh[send,h]
__global__ void k_edge_bwd(const int* __restrict__ eidx, const float* __restrict__ Ybuf,
                           const float* __restrict__ gm, const float* __restrict__ hbuf,
                           const float* __restrict__ Rbuf,
                           float* __restrict__ gR, float* __restrict__ gY,
                           float* __restrict__ gh, int use_gh)
{
    int e = (blockIdx.x * blockDim.x + threadIdx.x) >> 5;
    int lane = threadIdx.x & 31;
    if (e >= NEDGES) return;                      // wave-uniform
    int s = eidx[e], r = eidx[NEDGES + e];
    int h0 = lane, h1 = lane + 32;
    float hv0 = hbuf[s * HID + h0], hv1 = hbuf[s * HID + h1];
    float Rv0 = Rbuf[(size_t)e * HID + h0], Rv1 = Rbuf[(size_t)e * HID + h1];
    float yk[NSH];
    const float* Ye = Ybuf + (size_t)e * NSH;
#pragma unroll
    for (int k = 0; k < NSH; ++k) yk[k] = Ye[k];
    const float* gmr = gm + (size_t)r * HID * NSH;
    float dot0 = 0.f, dot1 = 0.f;
    float gy[NSH];
#pragma unroll
    for (int k = 0; k < NSH; ++k) {
        float a = gmr[h0 * NSH + k];
        float b = gmr[h1 * NSH + k];
        dot0 += a * yk[k];
        dot1 += b * yk[k];
        gy[k] = hv0 * Rv0 * a + hv1 * Rv1 * b;
    }
    gR[(size_t)e * HID + h0] = hv0 * dot0;
    gR[(size_t)e * HID + h1] = hv1 * dot1;
    if (use_gh) {
        atomicAdd(&gh[s * HID + h0], Rv0 * dot0);
        atomicAdd(&gh[s * HID + h1], Rv1 * dot1);
    }
#pragma unroll
    for (int k = 0; k < NSH; ++k) {
        float v = gy[k];
        for (int off = 16; off > 0; off >>= 1) v += __shfl_xor(v, off, 32);
        if (lane == 0) gY[(size_t)e * NSH + k] = v;
    }
}

__global__ void k_sc_bwd(const float* __restrict__ elem, const float* __restrict__ Wsc,
                         const float* __restrict__ gout, float* __restrict__ gin)
{
    int idx = blockIdx.x * blockDim.x + threadIdx.x;
    if (idx >= NNODES * HID) return;
    int n = idx >> 6, h = idx & 63;
    int z = (int)elem[n];
    const float* W = Wsc + (size_t)z * HID * HID + h * HID;
    const float* g = gout + (size_t)n * HID;
    float acc = 0.f;
    for (int k = 0; k < HID; ++k) acc += W[k] * g[k];
    gin[idx] = acc;
}

__global__ void k_silu_bwd(float* __restrict__ gT, const float* __restrict__ pre, int n)
{
    int i = blockIdx.x * blockDim.x + threadIdx.x;
    if (i < n) gT[i] *= silugradf_(pre[i]);
}

__global__ void k_add_ro0(float* __restrict__ g, const float* __restrict__ ro0)
{
    int idx = blockIdx.x * blockDim.x + threadIdx.x;
    if (idx >= NNODES * HID) return;
    g[idx] += ro0[idx & 63];
}

__global__ void k_geom_bwd(const float* __restrict__ pos, const float* __restrict__ shifts,
                           const int* __restrict__ eidx, const float* __restrict__ gY,
                           const float* __restrict__ gef, float* __restrict__ forces)
{
    int e = blockIdx.x * blockDim.x + threadIdx.x;
    if (e >= NEDGES) return;
    int s = eidx[e], r = eidx[NEDGES + e];
    float vx = pos[r * 3 + 0] - pos[s * 3 + 0] + shifts[e * 3 + 0];
    float vy = pos[r * 3 + 1] - pos[s * 3 + 1] + shifts[e * 3 + 1];
    float vz = pos[r * 3 + 2] - pos[s * 3 + 2] + shifts[e * 3 + 2];
    float rl = sqrtf(vx * vx + vy * vy + vz * vz);
    float rr = rl + 1e-9f;
    float inv = 1.f / rr;
    float x = vx * inv, y = vy * inv, z = vz * inv;
    float x2 = x * x, y2 = y * y, z2 = z * z;
    const float s3 = 1.7320508f;
    const float* gy_ = gY + (size_t)e * NSH;

    float gx = 0.f, gyv = 0.f, gz = 0.f;
    gx  += gy_[1];
    gyv += gy_[2];
    gz  += gy_[3];
    gx  += gy_[4] * s3 * y;   gyv += gy_[4] * s3 * x;
    gyv += gy_[5] * s3 * z;   gz  += gy_[5] * s3 * y;
    gz  += gy_[6] * 3.f * z;
    gx  += gy_[7] * s3 * z;   gz  += gy_[7] * s3 * x;
    gx  += gy_[8] * s3 * x;   gyv -= gy_[8] * s3 * y;
    gx  += gy_[9] * (4.7436f * x * y);
    gyv += gy_[9] * (2.3718f * (x2 - y2));
    gx  += gy_[10] * 3.873f * y * z;
    gyv += gy_[10] * 3.873f * x * z;
    gz  += gy_[10] * 3.873f * x * y;
    gx  += gy_[11] * (-1.2248f * x * y);
    gyv += gy_[11] * 0.6124f * (4.f * z2 - x2 - 3.f * y2);
    gz  += gy_[11] * (4.8992f * y * z);
    gx  += gy_[12] * (-3.f * x * z);
    gyv += gy_[12] * (-3.f * y * z);
    gz  += gy_[12] * (3.f * z2 - 1.5f * (x2 + y2));
    gx  += gy_[13] * 0.6124f * (4.f * z2 - 3.f * x2 - y2);
    gyv += gy_[13] * (-1.2248f * x * y);
    gz  += gy_[13] * (4.8992f * x * z);
    gx  += gy_[14] * (3.873f * x * z);
    gyv += gy_[14] * (-3.873f * y * z);
    gz  += gy_[14] * 1.9365f * (x2 - y2);
    gx  += gy_[15] * (2.3718f * (x2 - y2));
    gyv += gy_[15] * (-4.7436f * x * y);

    // Bessel/cutoff chain: g_r
    const float* ge = gef + (size_t)e * 16;       // only cols 0..7 valid
    float uc = fminf(rl * (1.f / RCUT), 1.f);
    float u2 = uc * uc, u4 = u2 * u2, u5 = u4 * uc, u6 = u5 * uc, u7 = u6 * uc, u8 = u7 * uc;
    float fc = 1.f - 28.f * u6 + 48.f * u7 - 21.f * u8;
    float omu = 1.f - uc;
    float dfc = -168.f * u5 * omu * omu * (1.f / RCUT);
    if (rl >= RCUT) { fc = 0.f; dfc = 0.f; }
    const float Cb = 0.63245553f;
    const float PI_ = 3.14159265358979f;
    float gr = 0.f;
#pragma unroll
    for (int nB = 1; nB <= NB; ++nB) {
        float a = (float)nB * PI_ * (1.f / RCUT);
        float sn = __sinf(a * rl), cn = __cosf(a * rl);
        float rb  = Cb * sn * inv;
        float drb = Cb * (a * cn * inv - sn * inv * inv);
        gr += ge[nB - 1] * (drb * fc + rb * dfc);
    }

    float gvx = 0.f, gvy = 0.f, gvz = 0.f;
    if (rl > 1e-12f) {
        float gu_v = gx * vx + gyv * vy + gz * vz;
        float c1 = 1.f / (rl * rr * rr);
        float c2 = gr / rl;
        gvx = gx  * inv - vx * gu_v * c1 + c2 * vx;
        gvy = gyv * inv - vy * gu_v * c1 + c2 * vy;
        gvz = gz  * inv - vz * gu_v * c1 + c2 * vz;
    }
    atomicAdd(&forces[r * 3 + 0], -gvx);
    atomicAdd(&forces[r * 3 + 1], -gvy);
    atomicAdd(&forces[r * 3 + 2], -gvz);
    atomicAdd(&forces[s * 3 + 0],  gvx);
    atomicAdd(&forces[s * 3 + 1],  gvy);
    atomicAdd(&forces[s * 3 + 2],  gvz);
}

// ---------------------------------------------------------------------------
// Host side
// ---------------------------------------------------------------------------
static inline int cdiv_(int a, int b) { return (a + b - 1) / b; }

template<int TRANSB, int ACT, int BETA, int NT, int INCA, int GUARDN, int KC>
static void gemm_t(hipStream_t st, const float* A, int lda, const float* B, int ldb,
                   float* C, int ldc, int M, int N)
{
    int tilesN = (N + 16 * NT - 1) / (16 * NT);
    int tiles = (M >> 4) * tilesN;
    k_gemm_t<TRANSB, ACT, BETA, NT, INCA, GUARDN, KC>
        <<<cdiv_(tiles, 8), 256, 0, st>>>(A, lda, B, ldb, C, ldc, M, N);
}

static void zero(hipStream_t st, float* p, long long n)
{
    k_zero<<<cdiv_((int)n, 256), 256, 0, st>>>(p, (int)n);
}

extern "C" void kernel_launch(void* const* d_in, const int* in_sizes, int n_in,
                              void* d_out, int out_size, void* d_ws, size_t ws_size,
                              hipStream_t stream)
{
    (void)in_sizes; (void)n_in; (void)out_size; (void)ws_size;
    // inputs in setup_inputs() dict order (nested dicts in insertion order)
    const float* positions = (const float*)d_in[0];
    const float* node_attrs = (const float*)d_in[1];
    const float* shifts = (const float*)d_in[2];
    const float* ae      = (const float*)d_in[3];
    const float* W_embed = (const float*)d_in[4];
    const float* W_ro0   = (const float*)d_in[5];
    const float* W_ro1a  = (const float*)d_in[6];
    const float* W_ro1b  = (const float*)d_in[7];
    const float* W_up[2]  = {(const float*)d_in[8],  (const float*)d_in[14]};
    const float* W_r1[2]  = {(const float*)d_in[9],  (const float*)d_in[15]};
    const float* W_r2[2]  = {(const float*)d_in[10], (const float*)d_in[16]};
    const float* W_sc[2]  = {(const float*)d_in[11], (const float*)d_in[17]};
    const float* W_p1[2]  = {(const float*)d_in[12], (const float*)d_in[18]};
    const float* W_p2[2]  = {(const float*)d_in[13], (const float*)d_in[19]};
    const int* eidx  = (const int*)d_in[20];
    const int* batch = (const int*)d_in[21];

    float* out = (float*)d_out;
    float* te      = out + OUT_TE;
    float* node_e  = out + OUT_NE;
    float* contrib = out + OUT_CONTRIB;
    float* forces  = out + OUT_F;
    float* nf_out  = out + OUT_NF;

    // workspace layout (floats)
    float* ws = (float*)d_ws;
    size_t o = 0;
    float* ef  = ws + o; o += (size_t)NEDGES * NB;
    float* Y   = ws + o; o += (size_t)NEDGES * NSH;
    float* T   = ws + o; o += (size_t)NEDGES * HID;       // radial scratch (t / pre)
    float* R[2]  = {ws + o, ws + o + (size_t)NEDGES * HID}; o += 2ull * NEDGES * HID;
    float* M_[2] = {ws + o, ws + o + (size_t)NNODES * HID * NSH}; o += 2ull * NNODES * HID * NSH;
    float* H[2]  = {ws + o, ws + o + (size_t)NNODES * HID}; o += 2ull * NNODES * HID;
    float* NF[3] = {ws + o, ws + o + (size_t)NNODES * HID, ws + o + 2ull * NNODES * HID};
    o += 3ull * NNODES * HID;
    float* Q   = ws + o; o += (size_t)NNODES * HID;
    float* P   = ws + o; o += (size_t)NNODES * 16;
    float* GP  = ws + o; o += (size_t)NNODES * 16;
    float* GA  = ws + o; o += (size_t)NNODES * HID;        // g wrt nf2
    float* GB  = ws + o; o += (size_t)NNODES * HID;        // g wrt nf1
    float* A1  = ws + o; o += (size_t)NNODES * HID;
    float* A2  = ws + o; o += (size_t)NNODES * HID;
    float* GM  = ws + o; o += (size_t)NNODES * HID * NSH;
    float* GH  = ws + o; o += (size_t)NNODES * HID;
    float* GR  = ws + o; o += (size_t)NEDGES * HID;
    float* GT  = ws + o; o += (size_t)NEDGES * HID;
    float* GEF = ws + o; o += (size_t)NEDGES * 16;
    float* GY  = ws + o; o += (size_t)NEDGES * NSH;
    float* ELEM = ws + o; o += (size_t)NNODES;

    // ---- init ----
    zero(stream, out, OUT_TOTAL);
    zero(stream, M_[0], (long long)NNODES * HID * NSH);
    zero(stream, M_[1], (long long)NNODES * HID * NSH);
    k_node_init<<<cdiv_(NNODES * HID, 256), 256, 0, stream>>>(node_attrs, ae, W_embed, batch,
                                                              NF[0], ELEM, node_e, contrib);
    k_edge_geom<<<cdiv_(NEDGES, 256), 256, 0, stream>>>(positions, shifts, eidx, ef, Y);

    // ---- forward layers ----
    for (int li = 0; li < 2; ++li) {
        const float* nfin = NF[li];
        float* nfo = NF[li + 1];
        // radial MLP: T = silu(ef@Wr1); R = T@Wr2
        gemm_t<0,1,0,4,1,0,NB >(stream, ef, NB, W_r1[li], HID, T, HID, NEDGES, HID);
        gemm_t<0,0,0,4,1,0,HID>(stream, T, HID, W_r2[li], HID, R[li], HID, NEDGES, HID);
        // h = nf_in @ W_up
        gemm_t<0,0,0,4,1,0,HID>(stream, nfin, HID, W_up[li], HID, H[li], HID, NNODES, HID);
        // scatter messages
        k_edge_msg<<<cdiv_(NEDGES * HID, 256), 256, 0, stream>>>(eidx, H[li], R[li], Y, M_[li]);
        // q = sum_k m^2 ; nf_out = sc + m0@Wp1 + q@Wp2
        k_node_q<<<cdiv_(NNODES * HID, 256), 256, 0, stream>>>(M_[li], Q);
        k_sc_fwd<<<cdiv_(NNODES * HID, 256), 256, 0, stream>>>(ELEM, W_sc[li], nfin, nfo);
        gemm_t<0,0,1,4,NSH,0,HID>(stream, M_[li], HID * NSH, W_p1[li], HID, nfo, HID, NNODES, HID);
        gemm_t<0,0,1,4,1,  0,HID>(stream, Q, HID, W_p2[li], HID, nfo, HID, NNODES, HID);
        // readouts
        if (li == 0) {
            k_readout0<<<cdiv_(NNODES, 256), 256, 0, stream>>>(nfo, W_ro0, batch, node_e, contrib);
        } else {
            gemm_t<0,0,0,1,1,0,HID>(stream, nfo, HID, W_ro1a, 16, P, 16, NNODES, 16);
            k_readout1<<<cdiv_(NNODES, 256), 256, 0, stream>>>(P, W_ro1b, batch, node_e, contrib);
        }
    }
    k_total<<<1, 64, 0, stream>>>(contrib, te);
    k_pack_nf<<<cdiv_(NNODES * 128, 256), 256, 0, stream>>>(NF[1], NF[2], nf_out);

    // ---- backward (forces) ----
    // readout1 -> g_nf2
    k_gp<<<cdiv_(NNODES * 16, 256), 256, 0, stream>>>(P, W_ro1b, GP);
    gemm_t<1,0,0,4,1,0,16>(stream, GP, 16, W_ro1a, 16, GA, HID, NNODES, HID);

    for (int li = 1; li >= 0; --li) {
        float* gout = (li == 1) ? GA : GB;
        // A1 = g@Wp1^T, A2 = g@Wp2^T
        gemm_t<1,0,0,4,1,0,HID>(stream, gout, HID, W_p1[li], HID, A1, HID, NNODES, HID);
        gemm_t<1,0,0,4,1,0,HID>(stream, gout, HID, W_p2[li], HID, A2, HID, NNODES, HID);
        k_gm<<<cdiv_(NNODES * HID, 256), 256, 0, stream>>>(A1, A2, M_[li], GM);
        if (li == 1) zero(stream, GH, (long long)NNODES * HID);
        k_edge_bwd<<<cdiv_(NEDGES * 32, 256), 256, 0, stream>>>(eidx, Y, GM, H[li], R[li],
                                                                GR, GY, GH, li == 1 ? 1 : 0);
        if (li == 1) {
            // grad wrt nf1: sc-path + (gh @ W_up^T)
            k_sc_bwd<<<cdiv_(NNODES * HID, 256), 256, 0, stream>>>(ELEM, W_sc[li], gout, GB);
            gemm_t<1,0,1,4,1,0,HID>(stream, GH, HID, W_up[li], HID, GB, HID, NNODES, HID);
        }
        // radial backward: pre = ef@Wr1 ; gT = gR@Wr2^T * silu'(pre) ; gef = gT@Wr1^T
        gemm_t<0,0,0,4,1,0,NB >(stream, ef, NB, W_r1[li], HID, T, HID, NEDGES, HID);
        gemm_t<1,0,0,4,1,0,HID>(stream, GR, HID, W_r2[li], HID, GT, HID, NEDGES, HID);
        k_silu_bwd<<<cdiv_(NEDGES * HID, 256), 256, 0, stream>>>(GT, T, NEDGES * HID);
        gemm_t<1,0,0,1,1,1,HID>(stream, GT, HID, W_r1[li], HID, GEF, 16, NEDGES, NB);
        // geometry backward -> forces
        k_geom_bwd<<<cdiv_(NEDGES, 256), 256, 0, stream>>>(positions, shifts, eidx, GY, GEF, forces);
        if (li == 1) {
            // add readout0 direct gradient before layer-0 backward
            k_add_ro0<<<cdiv_(NNODES * HID, 256), 256, 0, stream>>>(GB, W_ro0);
        }
    }
}